// ReasoningAttention_9139690405988
// MI455X (gfx1250) — compile-verified
//
#include <hip/hip_runtime.h>

typedef __attribute__((ext_vector_type(2))) float v2f;
typedef __attribute__((ext_vector_type(4))) float v4f;
typedef __attribute__((ext_vector_type(8))) float v8f;

#define NCONCEPTS 4096
#define CDIM      1024
#define BATCH     4096
#define KSAMP     256

// ---------------------------------------------------------------------------
// Kernel 1: s[c] = sum_d cb[c,d] * att[c,d]
//
// Strategy: per wave, own a 16-row tile. For each K-chunk of 4 columns, build
// the elementwise product tile P (16x4 f32, A-operand layout: lane m<16 holds
// k={0,1} of row m, lane m+16 holds k={2,3}) with one b64 load per operand,
// then issue V_WMMA_F32_16X16X4_F32 with B = all-ones 4x16. Each column of
// the 16x16 f32 accumulator is then the running row-sum; after the loop,
// lane 0 / lane 16 hold s for rows 0..7 / 8..15 of the tile in their 8
// accumulator VGPRs (C/D layout: VGPR r, lanes 0-15 -> M=r, lanes 16-31 ->
// M=r+8) and store them directly.
// ---------------------------------------------------------------------------
__global__ __launch_bounds__(256) void rowdot_wmma_f32(
    const float* __restrict__ cb,
    const float* __restrict__ att,
    float* __restrict__ s) {
  const int lane = threadIdx.x & 31;
  const int wave = threadIdx.x >> 5;
  const int tile = blockIdx.x * 8 + wave;   // 8 waves per block, 16 rows/tile
  const int row0 = tile * 16;
  const int m    = lane & 15;               // row within tile
  const int half = lane >> 4;               // 0 -> k={0,1}, 1 -> k={2,3}

  const float* cbp = cb  + (size_t)(row0 + m) * CDIM + 2 * half;
  const float* atp = att + (size_t)(row0 + m) * CDIM + 2 * half;

  const v2f ones = {1.0f, 1.0f};            // B-matrix: all ones (layout-invariant)
  v8f acc = {};

#pragma unroll 8
  for (int d = 0; d < CDIM; d += 4) {
    v2f a = *(const v2f*)(cbp + d);         // global_load_b64
    v2f w = *(const v2f*)(atp + d);         // global_load_b64
    v2f p = a * w;                          // P = cb .* att  (2 x v_mul_f32)
    // D = P(16x4) x ones(4x16) + C : every column accumulates the row sums.
    // 8 args: (neg_a, A, neg_b, B, c_mod, C, reuse_a, reuse_b)
    acc = __builtin_amdgcn_wmma_f32_16x16x4_f32(
        /*neg_a=*/false, p, /*neg_b=*/false, ones,
        /*c_mod=*/(short)0, acc, /*reuse_a=*/false, /*reuse_b=*/false);
  }

  // Column N=0 of D: lane 0 VGPRs 0..7 = s[row0+0..7],
  //                  lane 16 VGPRs 0..7 = s[row0+8..15].
  if (m == 0) {
    *(v8f*)(s + row0 + 8 * half) = acc;     // 2x global_store_b128
  }
}

// ---------------------------------------------------------------------------
// Kernel 2: out[b, :] = 0; out[b, idx[b,k]] = s[idx[b,k]]
// One block per batch row; duplicates in idx are idempotent writes.
// ---------------------------------------------------------------------------
__global__ __launch_bounds__(256) void scatter_rows(
    const float* __restrict__ s,
    const int* __restrict__ idx,
    float* __restrict__ out) {
  const int b = blockIdx.x;
  float* row = out + (size_t)b * NCONCEPTS;

  const v4f z = {};
#pragma unroll
  for (int j = threadIdx.x; j < NCONCEPTS / 4; j += 256) {
    ((v4f*)row)[j] = z;                     // global_store_b128, 16 KB/row
  }
  __syncthreads();

  const int c = idx[(size_t)b * KSAMP + threadIdx.x];
  row[c] = s[c];
}

extern "C" void kernel_launch(void* const* d_in, const int* in_sizes, int n_in,
                              void* d_out, int out_size, void* d_ws, size_t ws_size,
                              hipStream_t stream) {
  const float* cb  = (const float*)d_in[0];   // [4096, 1024] f32
  const float* att = (const float*)d_in[1];   // [4096, 1024] f32
  const int*   idx = (const int*)d_in[2];     // [4096, 256]  int
  float* out = (float*)d_out;                 // [4096, 4096] f32
  float* s   = (float*)d_ws;                  // [4096] f32 scratch (16 KB)

  // 8 tiles (128 rows) per block -> 32 blocks.
  rowdot_wmma_f32<<<NCONCEPTS / 128, 256, 0, stream>>>(cb, att, s);
  // One block per batch row.
  scatter_rows<<<BATCH, 256, 0, stream>>>(s, idx, out);
}